// AttentionBlock_26757646254386
// MI455X (gfx1250) — compile-verified
//
#include <hip/hip_runtime.h>
#include <hip/hip_bf16.h>
#include <math.h>

// Problem constants (from reference)
#define BATCH   16
#define NIDX    128
#define DIM     128
#define LATENT  512
#define FFHID   512
#define ROWS    (BATCH * NIDX)   // 2048
#define EPSLN   1e-5f

typedef __attribute__((ext_vector_type(2))) float v2f;
typedef __attribute__((ext_vector_type(8))) float v8f;

__device__ __forceinline__ v8f wmma4(v2f a, v2f b, v8f c) {
    return __builtin_amdgcn_wmma_f32_16x16x4_f32(
        /*neg_a=*/false, a, /*neg_b=*/false, b,
        /*c_mod=*/(short)0, c, /*reuse_a=*/false, /*reuse_b=*/false);
}

// ---------------------------------------------------------------------------
// fp32 WMMA GEMM: Out[M,N] = A[M,K] @ B[K,N] + bias[N].
// One wave computes a 16x64 tile: 4 accumulators share one A fragment per
// k-step -> 4 back-to-back v_wmma_f32_16x16x4_f32 per A load.
// Compile-time M/N/K so all strides fold to immediates.
// MODE 0: store (acc+bias). MODE 1: store relu(acc+bias).
// MODE 2: s = tanh(acc+bias)*wvec[col]; row-sum over 64 cols via shfl_xor and
//         atomicAdd into rowsum[row] (Out unused).
// ---------------------------------------------------------------------------
template <int MODE, int M, int N, int K>
__global__ void wmma_gemm_f32(const float* __restrict__ A,
                              const float* __restrict__ Bm,
                              const float* __restrict__ bias,
                              float* __restrict__ Out,
                              const float* __restrict__ wvec,
                              float* __restrict__ rowsum) {
    const int wave = (blockIdx.x * blockDim.x + threadIdx.x) >> 5;
    const int lane = threadIdx.x & 31;
    const int half = lane >> 4;     // 0: K{0,1}/rows 0-7, 1: K{2,3}/rows 8-15
    const int l16  = lane & 15;

    constexpr int NTN = N / 64;                 // 64-wide col tiles
    const int row0 = (wave / NTN) * 16;
    const int col0 = (wave % NTN) * 64;

    const float* __restrict__ Arow  = A + (size_t)(row0 + l16) * K;
    const float* __restrict__ Bbase = Bm + col0 + l16;

    v8f acc[4] = {v8f{}, v8f{}, v8f{}, v8f{}};

#pragma unroll 2
    for (int k = 0; k < K; k += 4) {
        // A fragment: row = row0+l16, K = k+2*half .. +1 (contiguous pair)
        const v2f a = *(const v2f*)(Arow + k + 2 * half);
        // B fragments: cols col0+16c+l16, K rows k+2*half and k+2*half+1
        const float* __restrict__ Br0 = Bbase + (size_t)(k + 2 * half) * N;
        const float* __restrict__ Br1 = Br0 + N;
        v2f b0, b1, b2, b3;
        b0.x = Br0[0];  b0.y = Br1[0];
        b1.x = Br0[16]; b1.y = Br1[16];
        b2.x = Br0[32]; b2.y = Br1[32];
        b3.x = Br0[48]; b3.y = Br1[48];
        acc[0] = wmma4(a, b0, acc[0]);
        acc[1] = wmma4(a, b1, acc[1]);
        acc[2] = wmma4(a, b2, acc[2]);
        acc[3] = wmma4(a, b3, acc[3]);
    }

    if (MODE == 2) {
        float rs[8] = {0.f, 0.f, 0.f, 0.f, 0.f, 0.f, 0.f, 0.f};
#pragma unroll
        for (int c = 0; c < 4; ++c) {
            const int col = col0 + 16 * c + l16;
            const float bs = bias[col];
            const float wv = wvec[col];
#pragma unroll
            for (int v = 0; v < 8; ++v)
                rs[v] += tanhf(acc[c][v] + bs) * wv;
        }
#pragma unroll
        for (int v = 0; v < 8; ++v) {
            float s = rs[v];
            // sum across the 16 lanes holding this tile-row's columns
            s += __shfl_xor(s, 8, 16);
            s += __shfl_xor(s, 4, 16);
            s += __shfl_xor(s, 2, 16);
            s += __shfl_xor(s, 1, 16);
            if (l16 == 0)
                atomicAdd(&rowsum[row0 + v + 8 * half], s);
        }
    } else {
#pragma unroll
        for (int c = 0; c < 4; ++c) {
            const int col = col0 + 16 * c + l16;
            const float bs = bias[col];
#pragma unroll
            for (int v = 0; v < 8; ++v) {
                float val = acc[c][v] + bs;
                if (MODE == 1) val = fmaxf(val, 0.0f);
                Out[(size_t)(row0 + v + 8 * half) * N + col] = val;
            }
        }
    }
}

// ---------------------------------------------------------------------------
// Per-batch: finish c[b,j] (add tanh(1)*wv_w[2L+N+j] eye term), softmax over
// j (shift-invariance already cancelled the q path and wv_b), then
// v[b,d] = sum_j w_j * in_obs[b,j,d].  Block = 128 threads (j == d == tid).
// ---------------------------------------------------------------------------
__global__ void softmax_attn(const float* __restrict__ cvec,
                             const float* __restrict__ wv_w,
                             const float* __restrict__ in_obs,
                             float* __restrict__ vout) {
    const int b = blockIdx.x;
    const int t = threadIdx.x;                  // 0..127
    __shared__ float wsm[NIDX];
    __shared__ float red[NIDX];

    const float tanh1 = tanhf(1.0f);
    float cj = cvec[b * NIDX + t] + tanh1 * wv_w[2 * LATENT + NIDX + t];

    red[t] = cj;
    __syncthreads();
    for (int s = 64; s > 0; s >>= 1) {
        if (t < s) red[t] = fmaxf(red[t], red[t + s]);
        __syncthreads();
    }
    const float mx = red[0];
    __syncthreads();

    const float e = __expf(cj - mx);
    wsm[t] = e;
    red[t] = e;
    __syncthreads();
    for (int s = 64; s > 0; s >>= 1) {
        if (t < s) red[t] += red[t + s];
        __syncthreads();
    }
    const float inv = 1.0f / red[0];
    __syncthreads();

    // v[b, d=t] = sum_j w_j * in_obs[b, j, d]   (coalesced over d)
    const float* __restrict__ base = in_obs + (size_t)b * NIDX * DIM;
    float acc = 0.0f;
#pragma unroll 4
    for (int j = 0; j < NIDX; ++j)
        acc = fmaf(wsm[j], base[(size_t)j * DIM + t], acc);
    vout[b * DIM + t] = acc * inv;
}

// ---------------------------------------------------------------------------
// out[row,:] = LayerNorm(x[row,:] + add, g, b).  add is either per-row
// [ROWS,D] (bcast==0) or per-batch [B,D] broadcast over N (bcast==1).
// Block = 128 threads, one row per block.
// ---------------------------------------------------------------------------
__global__ void layernorm_add(const float* __restrict__ x,
                              const float* __restrict__ add,
                              int bcast,
                              const float* __restrict__ g,
                              const float* __restrict__ bb,
                              float* __restrict__ out) {
    const int row = blockIdx.x;
    const int t = threadIdx.x;                  // 0..127
    const int b = row >> 7;                     // row / NIDX
    __shared__ float red[DIM];

    const float xa = x[(size_t)row * DIM + t];
    const float ad = bcast ? add[b * DIM + t] : add[(size_t)row * DIM + t];
    const float val = xa + ad;

    red[t] = val;
    __syncthreads();
    for (int s = 64; s > 0; s >>= 1) {
        if (t < s) red[t] += red[t + s];
        __syncthreads();
    }
    const float mu = red[0] * (1.0f / DIM);
    __syncthreads();

    const float d = val - mu;
    red[t] = d * d;
    __syncthreads();
    for (int s = 64; s > 0; s >>= 1) {
        if (t < s) red[t] += red[t + s];
        __syncthreads();
    }
    const float var = red[0] * (1.0f / DIM);

    out[(size_t)row * DIM + t] = d * rsqrtf(var + EPSLN) * g[t] + bb[t];
}

// ---------------------------------------------------------------------------
extern "C" void kernel_launch(void* const* d_in, const int* in_sizes, int n_in,
                              void* d_out, int out_size, void* d_ws, size_t ws_size,
                              hipStream_t stream) {
    const float* in_obs = (const float*)d_in[0];
    // d_in[1], d_in[2] (Wq_w, Wq_b) and d_in[6] (wv_b) cancel exactly under
    // softmax shift-invariance -- unused.
    const float* Wk_w  = (const float*)d_in[3];
    const float* Wk_b  = (const float*)d_in[4];
    const float* wv_w  = (const float*)d_in[5];
    const float* g1    = (const float*)d_in[7];
    const float* b1    = (const float*)d_in[8];
    const float* ff1_w = (const float*)d_in[9];
    const float* ff1_b = (const float*)d_in[10];
    const float* ff2_w = (const float*)d_in[11];
    const float* ff2_b = (const float*)d_in[12];
    const float* g2    = (const float*)d_in[13];
    const float* b2    = (const float*)d_in[14];
    float* out = (float*)d_out;

    // Workspace layout (floats): cvec[2048] | v[2048] | res[2048*128] |
    // H[2048*512] | F[2048*128]  -> ~6.0 MB
    float* ws   = (float*)d_ws;
    float* cvec = ws;                              // ROWS
    float* vbuf = ws + 2048;                       // B*D (padded slot)
    float* resb = ws + 4096;                       // ROWS*DIM
    float* Hbuf = resb + (size_t)ROWS * DIM;       // ROWS*FFHID
    float* Fbuf = Hbuf + (size_t)ROWS * FFHID;     // ROWS*DIM

    // c accumulated with atomics -> must start at zero every call
    hipMemsetAsync(cvec, 0, ROWS * sizeof(float), stream);

    // 1) K-projection GEMM [2048,128]@[128,512], fused tanh*w row-reduce -> c
    //    16x64 tiles: waves = 128*8 = 1024, 8 waves/block -> 128 blocks
    wmma_gemm_f32<2, ROWS, LATENT, DIM><<<128, 256, 0, stream>>>(
        in_obs, Wk_w, Wk_b, nullptr, wv_w + (LATENT + NIDX), cvec);

    // 2) softmax over j + weighted row-sum -> v[b,:]
    softmax_attn<<<BATCH, NIDX, 0, stream>>>(cvec, wv_w, in_obs, vbuf);

    // 3) res = LN(in_obs + v_broadcast; g1,b1)
    layernorm_add<<<ROWS, DIM, 0, stream>>>(in_obs, vbuf, 1, g1, b1, resb);

    // 4) H = relu(res @ ff1_w + ff1_b)   [2048,128]@[128,512]
    wmma_gemm_f32<1, ROWS, FFHID, DIM><<<128, 256, 0, stream>>>(
        resb, ff1_w, ff1_b, Hbuf, nullptr, nullptr);

    // 5) F = H @ ff2_w + ff2_b           [2048,512]@[512,128]
    //    16x64 tiles: waves = 128*2 = 256 -> 32 blocks
    wmma_gemm_f32<0, ROWS, DIM, FFHID><<<32, 256, 0, stream>>>(
        Hbuf, ff2_w, ff2_b, Fbuf, nullptr, nullptr);

    // 6) out = LN(F + in_obs; g2,b2)
    layernorm_add<<<ROWS, DIM, 0, stream>>>(Fbuf, in_obs, 0, g2, b2, out);
}